// SRenderY_43087111914144
// MI455X (gfx1250) — compile-verified
//
#include <hip/hip_runtime.h>
#include <hip/hip_bf16.h>
#include <math.h>

#define B_  8
#define V_  5023
#define F_  9976
#define H_  512
#define W_  512
#define UV_ 256

typedef __attribute__((ext_vector_type(2))) float v2f;
typedef __attribute__((ext_vector_type(8))) float v8f;

// ---------------------------------------------------------------------------
// Kernel 1: zero the normal accumulator (d_ws scratch; must be deterministic)
// ---------------------------------------------------------------------------
__global__ void k_zero(float* __restrict__ acc, int n) {
    int i = blockIdx.x * blockDim.x + threadIdx.x;
    if (i < n) acc[i] = 0.0f;
}

// ---------------------------------------------------------------------------
// Kernel 2: per-face cross products, atomically scatter-added per vertex
// ---------------------------------------------------------------------------
__global__ void k_accum_normals(const float* __restrict__ verts,
                                const int*   __restrict__ faces,
                                float* __restrict__ acc) {
    int t = blockIdx.x * blockDim.x + threadIdx.x;
    if (t >= B_ * F_) return;
    int b = t / F_;
    int f = t - b * F_;
    int i0 = faces[f * 3 + 0], i1 = faces[f * 3 + 1], i2 = faces[f * 3 + 2];
    const float* vb = verts + (size_t)b * V_ * 3;
    float v0x = vb[i0*3+0], v0y = vb[i0*3+1], v0z = vb[i0*3+2];
    float v1x = vb[i1*3+0], v1y = vb[i1*3+1], v1z = vb[i1*3+2];
    float v2x = vb[i2*3+0], v2y = vb[i2*3+1], v2z = vb[i2*3+2];

    // n0 = cross(v1-v0, v2-v0); n1 = cross(v2-v1, v0-v1); n2 = cross(v0-v2, v1-v2)
    float ax, ay, az, bx, by, bz;
    ax=v1x-v0x; ay=v1y-v0y; az=v1z-v0z; bx=v2x-v0x; by=v2y-v0y; bz=v2z-v0z;
    float n0x = ay*bz - az*by, n0y = az*bx - ax*bz, n0z = ax*by - ay*bx;
    ax=v2x-v1x; ay=v2y-v1y; az=v2z-v1z; bx=v0x-v1x; by=v0y-v1y; bz=v0z-v1z;
    float n1x = ay*bz - az*by, n1y = az*bx - ax*bz, n1z = ax*by - ay*bx;
    ax=v0x-v2x; ay=v0y-v2y; az=v0z-v2z; bx=v1x-v2x; by=v1y-v2y; bz=v1z-v2z;
    float n2x = ay*bz - az*by, n2y = az*bx - ax*bz, n2z = ax*by - ay*bx;

    float* ab = acc + (size_t)b * V_ * 3;
    atomicAdd(&ab[i0*3+0], n0x); atomicAdd(&ab[i0*3+1], n0y); atomicAdd(&ab[i0*3+2], n0z);
    atomicAdd(&ab[i1*3+0], n1x); atomicAdd(&ab[i1*3+1], n1y); atomicAdd(&ab[i1*3+2], n1z);
    atomicAdd(&ab[i2*3+0], n2x); atomicAdd(&ab[i2*3+1], n2y); atomicAdd(&ab[i2*3+2], n2z);
}

// ---------------------------------------------------------------------------
// Kernel 3: normalize accumulated normals in place
// ---------------------------------------------------------------------------
__global__ void k_normalize(float* __restrict__ acc, int n) {
    int i = blockIdx.x * blockDim.x + threadIdx.x;
    if (i >= n) return;
    float x = acc[i*3+0], y = acc[i*3+1], z = acc[i*3+2];
    float nrm = sqrtf(x*x + y*y + z*z);
    float inv = 1.0f / fmaxf(nrm, 1e-6f);
    acc[i*3+0] = x*inv; acc[i*3+1] = y*inv; acc[i*3+2] = z*inv;
}

// ---------------------------------------------------------------------------
// Kernel 4: fused rasterize-interpolate + bilinear albedo sample + SH shading.
// SH einsum (9x3 coeff shared per batch) mapped to V_WMMA_F32_16X16X4_F32:
// per wave: 32 pixels = 2 tiles of (16 pixels x 12-padded-K x 16 channels),
// K=9 accumulated as 3 chunks of 4.
// ---------------------------------------------------------------------------
#define WAVE_SH_STRIDE 10                 // per-pixel SH stride in LDS (conflict-free)
#define WAVE_REGION    (32*WAVE_SH_STRIDE + 512)   // sh stage + 2x(16x16) D stage
__global__ __launch_bounds__(256)
void k_render(const float* __restrict__ albedos,
              const float* __restrict__ lights,
              const float* __restrict__ uvc,
              const int*   __restrict__ faces,
              const int*   __restrict__ p2f_arr,
              const float* __restrict__ bary,
              const float* __restrict__ nrm,
              float* __restrict__ out) {
    __shared__ float lds[32 + 8 * WAVE_REGION];

    const int tid  = threadIdx.x;
    const int lane = tid & 31;
    const int wv   = tid >> 5;
    const int b    = blockIdx.x >> 10;                    // 1024 blocks / image
    const int pp   = ((blockIdx.x & 1023) << 8) | tid;    // pixel index in image
    const int h    = pp >> 9;
    const int w    = pp & 511;

    if (tid < 27) lds[tid] = lights[b * 27 + tid];        // 9x3 SH coeffs, this batch

    // ---- phase 1: per-pixel gather + interpolate + bilinear sample ----
    const int   pix = (b * H_ + h) * W_ + w;
    const int   p2f = p2f_arr[pix];
    const float b0 = bary[pix*3+0], b1 = bary[pix*3+1], b2 = bary[pix*3+2];

    float nx = 0.f, ny = 0.f, nz = 0.f;
    float a0 = 0.f, a1 = 0.f, a2 = 0.f;
    float alpha = 0.f;

    if (p2f >= 0) {
        alpha = 1.0f;
        unsigned u  = (unsigned)p2f;
        unsigned bb = u / (unsigned)F_;       // attrs_flat index spans B*F
        unsigned ff = u - bb * (unsigned)F_;
        int i0 = faces[ff*3+0], i1 = faces[ff*3+1], i2 = faces[ff*3+2];

        const float* uvb = uvc + (size_t)ff * 9;          // (F,3,3): u,v,1
        float gx = b0*uvb[0] + b1*uvb[3] + b2*uvb[6];
        float gy = b0*uvb[1] + b1*uvb[4] + b2*uvb[7];

        const float* nb = nrm + (size_t)bb * V_ * 3;
        nx = b0*nb[i0*3+0] + b1*nb[i1*3+0] + b2*nb[i2*3+0];
        ny = b0*nb[i0*3+1] + b1*nb[i1*3+1] + b2*nb[i2*3+1];
        nz = b0*nb[i0*3+2] + b1*nb[i1*3+2] + b2*nb[i2*3+2];

        // bilinear grid sample with zero border (matches reference tap())
        float x   = (gx + 1.0f) * (UV_ * 0.5f) - 0.5f;
        float y   = (gy + 1.0f) * (UV_ * 0.5f) - 0.5f;
        float x0f = floorf(x), y0f = floorf(y);
        float wx1 = x - x0f,  wy1 = y - y0f;
        float wx0 = 1.0f - wx1, wy0 = 1.0f - wy1;
        const float* ab = albedos + (size_t)b * 3 * UV_ * UV_;
        #pragma unroll
        for (int dy = 0; dy < 2; ++dy) {
            #pragma unroll
            for (int dx = 0; dx < 2; ++dx) {
                float ixf = x0f + (float)dx, iyf = y0f + (float)dy;
                bool valid = (ixf >= 0.f) & (ixf < (float)UV_) &
                             (iyf >= 0.f) & (iyf < (float)UV_);
                float wt = (dx ? wx1 : wx0) * (dy ? wy1 : wy0);
                if (valid) {
                    int o = (int)iyf * UV_ + (int)ixf;
                    a0 += wt * ab[o];
                    a1 += wt * ab[UV_*UV_ + o];
                    a2 += wt * ab[2*UV_*UV_ + o];
                }
            }
        }
    }

    // SH basis (reference's exact constant-factor ordering)
    float* shw = lds + 32 + wv * WAVE_REGION;
    shw[lane*WAVE_SH_STRIDE + 0] = 0.8862269254f;
    shw[lane*WAVE_SH_STRIDE + 1] = 1.0233267f * nx;
    shw[lane*WAVE_SH_STRIDE + 2] = 1.0233267f * ny;
    shw[lane*WAVE_SH_STRIDE + 3] = 1.0233267f * nz;
    shw[lane*WAVE_SH_STRIDE + 4] = 0.8580855f * nx * ny;
    shw[lane*WAVE_SH_STRIDE + 5] = 0.8580855f * nx * nz;
    shw[lane*WAVE_SH_STRIDE + 6] = 0.2477080f * ny * nz;
    shw[lane*WAVE_SH_STRIDE + 7] = 0.8580855f * (nx*nx - ny*ny);
    shw[lane*WAVE_SH_STRIDE + 8] = 0.8580855f * (3.0f*nz*nz - 1.0f);
    __syncthreads();

    // ---- phase 2: WMMA shading = SH(16x12pad) x coeff(12pad x 16) ----
    float* dD = lds + 32 + wv * WAVE_REGION + 32 * WAVE_SH_STRIDE;
    const int m    = lane & 15;
    const int hi   = lane >> 4;      // A layout: lanes 0-15 hold K={0,1}, 16-31 K={2,3}
    const int koff = hi * 2;

    auto coeff = [&](int s) -> float {       // B element: row s, col m (3 live cols)
        return (s < 9 && m < 3) ? lds[s * 3 + m] : 0.0f;
    };
    v2f B0, B1, B2;
    B0.x = coeff(koff + 0); B0.y = coeff(koff + 1);
    B1.x = coeff(koff + 4); B1.y = coeff(koff + 5);
    B2.x = coeff(koff + 8); B2.y = 0.0f;

    #pragma unroll
    for (int t = 0; t < 2; ++t) {            // two 16-pixel tiles per wave
        int pr = (t * 16 + m) * WAVE_SH_STRIDE;
        v2f A0, A1, A2;
        A0.x = shw[pr + koff + 0]; A0.y = shw[pr + koff + 1];
        A1.x = shw[pr + koff + 4]; A1.y = shw[pr + koff + 5];
        A2.x = hi ? 0.0f : shw[pr + 8];
        A2.y = 0.0f;
        v8f cacc = {0.f, 0.f, 0.f, 0.f, 0.f, 0.f, 0.f, 0.f};
        cacc = __builtin_amdgcn_wmma_f32_16x16x4_f32(false, A0, false, B0, (short)0, cacc, false, false);
        cacc = __builtin_amdgcn_wmma_f32_16x16x4_f32(false, A1, false, B1, (short)0, cacc, false, false);
        cacc = __builtin_amdgcn_wmma_f32_16x16x4_f32(false, A2, false, B2, (short)0, cacc, false, false);
        // D layout: VGPR r -> row (r + hi*8), col m
        #pragma unroll
        for (int r = 0; r < 8; ++r)
            dD[t * 256 + (r + hi * 8) * 16 + m] = cacc[r];
    }
    __syncthreads();

    // ---- phase 3: images = albedo * shading * alpha ----
    {
        int tt = lane >> 4, mm = lane & 15;  // this lane's pixel = tile tt, row mm
        float s0 = dD[tt * 256 + mm * 16 + 0];
        float s1 = dD[tt * 256 + mm * 16 + 1];
        float s2 = dD[tt * 256 + mm * 16 + 2];
        size_t plane = (size_t)H_ * W_;
        size_t ob = (size_t)b * 3 * plane + (size_t)h * W_ + w;
        out[ob]             = a0 * s0 * alpha;
        out[ob + plane]     = a1 * s1 * alpha;
        out[ob + 2 * plane] = a2 * s2 * alpha;
    }
}

// ---------------------------------------------------------------------------
extern "C" void kernel_launch(void* const* d_in, const int* in_sizes, int n_in,
                              void* d_out, int out_size, void* d_ws, size_t ws_size,
                              hipStream_t stream) {
    const float* vertices = (const float*)d_in[0];
    // d_in[1] = transformed_vertices: dead in the forward output, skipped.
    const float* albedos  = (const float*)d_in[2];
    const float* lights   = (const float*)d_in[3];
    const float* uvc      = (const float*)d_in[4];
    const int*   faces    = (const int*)d_in[5];
    const int*   p2f      = (const int*)d_in[6];
    const float* bary     = (const float*)d_in[7];
    float*       out      = (float*)d_out;
    float*       acc      = (float*)d_ws;           // B*V*3 floats (482 KB)

    const int nacc = B_ * V_ * 3;
    k_zero<<<(nacc + 255) / 256, 256, 0, stream>>>(acc, nacc);
    k_accum_normals<<<(B_ * F_ + 255) / 256, 256, 0, stream>>>(vertices, faces, acc);
    k_normalize<<<(B_ * V_ + 255) / 256, 256, 0, stream>>>(acc, B_ * V_);
    k_render<<<B_ * 1024, 256, 0, stream>>>(albedos, lights, uvc, faces, p2f, bary, acc, out);
}